// StateNeuronSep_8074538517095
// MI455X (gfx1250) — compile-verified
//
#include <hip/hip_runtime.h>
#include <math.h>

// ---------------------------------------------------------------------------
// MI455X observer recurrence:
//   y_t = C h_t + D u_t ;  h_{t+1} = A h_t + B u_t + L (yobs_t - y_t)
// Persistent kernel, A (64MB, L2-resident) streamed per step, h_t in LDS,
// matvec via V_WMMA_F32_16X16X4_F32 (fp32 for 4096-step numerical stability),
// one device-wide sense-reversal barrier per step, deterministic fixed-order
// reductions (no float atomics).
// ---------------------------------------------------------------------------

#define ORDER    4096
#define IN_DIM   8
#define SEQ_LEN  4096
#define NWG      128
#define NTHREADS 256
#define ROWS_PER_WG (ORDER / NWG)     // 32 rows per workgroup
#define KSLICE      (ORDER / 4)       // 1024 K per wave (4 K-slices)

typedef __attribute__((ext_vector_type(2))) float v2f;
typedef __attribute__((ext_vector_type(8))) float v8f;

struct Ws {
    float h0[ORDER];          // h ping
    float h1[ORDER];          // h pong
    float yp0[NWG];           // per-WG partials of C·h (ping)
    float yp1[NWG];           // per-WG partials of C·h (pong)
    float ybuf[SEQ_LEN];      // C·h_t per step (for final tanh kernel)
    float g[SEQ_LEN];         // D·u_t precomputed
    unsigned cnt;             // global barrier arrive counter
    unsigned gen;             // global barrier generation
};

__global__ void setup_kernel(const float* __restrict__ u,
                             const float* __restrict__ wd, Ws* ws) {
    int i = blockIdx.x * blockDim.x + threadIdx.x;
    if (i < ORDER) { ws->h0[i] = 1.0f; ws->h1[i] = 0.0f; }   // h0 = ones
    if (i < SEQ_LEN) {
        float gt = 0.0f;
        #pragma unroll
        for (int j = 0; j < IN_DIM; ++j) gt += wd[j] * u[(size_t)j * SEQ_LEN + i];
        ws->g[i]    = gt;
        ws->ybuf[i] = 0.0f;
    }
    if (i < NWG) { ws->yp0[i] = 0.0f; ws->yp1[i] = 0.0f; }
    if (i == 0)  { ws->cnt = 0u; ws->gen = 0u; }
}

// Deterministic single-block reduction: yp0[0] = sum(C) = C·h_0 (h_0 = ones).
__global__ void c_sum_kernel(const float* __restrict__ wc, Ws* ws) {
    __shared__ float red[NTHREADS];
    int tid = threadIdx.x;
    float v = 0.0f;
    for (int j = tid; j < ORDER; j += NTHREADS) v += wc[j];
    red[tid] = v;
    __syncthreads();
    for (int s = NTHREADS / 2; s > 0; s >>= 1) {
        if (tid < s) red[tid] += red[tid + s];
        __syncthreads();
    }
    if (tid == 0) ws->yp0[0] = red[0];
}

__global__ __launch_bounds__(NTHREADS)
void observer_kernel(const float* __restrict__ A, const float* __restrict__ B,
                     const float* __restrict__ C, const float* __restrict__ L,
                     const float* __restrict__ u, const float* __restrict__ yobs,
                     Ws* __restrict__ ws) {
    __shared__ float sh[ORDER];           // h_t staged in LDS (16 KB of 320 KB)
    __shared__ float spart[2 * 4 * 16];   // [rowgroup][kslice][m] partial sums
    __shared__ float s_y;                 // broadcast of C·h_t

    const int tid  = threadIdx.x;
    const int wg   = blockIdx.x;
    const int lane = tid & 31;
    const int wv   = tid >> 5;            // 0..7 (8 wave32s)
    const int rg   = wv & 1;              // row-group (2 x 16 rows)
    const int kq   = wv >> 1;             // K-slice (4 x 1024)
    const int half = lane >> 4;           // 0: K%4 in {0,1}; 1: {2,3}
    const int lrow = lane & 15;           // M within 16-row tile
    const int rowbase = wg * ROWS_PER_WG + rg * 16;
    const int k0 = kq * KSLICE;

    const float* __restrict__ Abase =
        A + (size_t)(rowbase + lrow) * ORDER + (size_t)(half * 2) + k0;

    float*    hcur = ws->h0;
    float*    hnxt = ws->h1;
    float*    ypc  = ws->yp0;
    float*    ypn  = ws->yp1;
    unsigned* cnt  = &ws->cnt;
    unsigned* gen  = &ws->gen;

    for (int t = 0; t < SEQ_LEN; ++t) {
        // ---- stage h_t into LDS (coalesced) -------------------------------
        for (int i = tid; i < ORDER; i += NTHREADS) sh[i] = hcur[i];

        // ---- deterministic fixed-tree reduce of per-WG C·h_t partials -----
        if (tid < 32) {
            float v = __hip_atomic_load(&ypc[tid],      __ATOMIC_RELAXED, __HIP_MEMORY_SCOPE_AGENT)
                    + __hip_atomic_load(&ypc[tid + 32], __ATOMIC_RELAXED, __HIP_MEMORY_SCOPE_AGENT)
                    + __hip_atomic_load(&ypc[tid + 64], __ATOMIC_RELAXED, __HIP_MEMORY_SCOPE_AGENT)
                    + __hip_atomic_load(&ypc[tid + 96], __ATOMIC_RELAXED, __HIP_MEMORY_SCOPE_AGENT);
            #pragma unroll
            for (int o = 16; o > 0; o >>= 1) v += __shfl_down(v, o, 32);
            if (tid == 0) {
                s_y = v;
                if (wg == 0) ws->ybuf[t] = v;   // y_t - D u_t, for output pass
            }
        }
        __syncthreads();

        // ---- 16-row x 1024-K slab of A h_t via V_WMMA_F32_16X16X4_F32 -----
        // A tile 16x4: lanes 0-15 hold K {0,1}, lanes 16-31 hold K {2,3}.
        // B 4x16: only column N=0 is h[k..k+3]; D column 0 = row dots.
        v8f acc = {};
        const float* ap = Abase;
        #pragma unroll 4
        for (int k = k0; k < k0 + KSLICE; k += 4, ap += 4) {
            v2f a = *(const v2f*)ap;                       // 8B-aligned b64 load
            v2f hv = *(const v2f*)&sh[k + half * 2];       // LDS b64 broadcast
            v2f b;
            b.x = (lrow == 0) ? hv.x : 0.0f;
            b.y = (lrow == 0) ? hv.y : 0.0f;
            acc = __builtin_amdgcn_wmma_f32_16x16x4_f32(
                      false, a, false, b, (short)0, acc, false, false);
        }
        // D column 0: lane 0 holds M=0..7 in acc[0..7], lane 16 holds M=8..15
        if (lrow == 0) {
            #pragma unroll
            for (int m = 0; m < 8; ++m)
                spart[(rg * 4 + kq) * 16 + half * 8 + m] = acc[m];
        }
        __syncthreads();

        // ---- finish 32 rows: combine K-slices, +B u_t, +L e_t; emit C·h --
        if (tid < ROWS_PER_WG) {
            const int trg = tid >> 4, m = tid & 15;
            const int row = wg * ROWS_PER_WG + tid;
            float s = 0.0f;
            #pragma unroll
            for (int q = 0; q < 4; ++q) s += spart[(trg * 4 + q) * 16 + m];
            float bu = 0.0f;
            #pragma unroll
            for (int j = 0; j < IN_DIM; ++j)
                bu += B[(size_t)row * IN_DIM + j] * u[(size_t)j * SEQ_LEN + t];
            const float e  = yobs[t] - (s_y + ws->g[t]);   // yobs_t - y_t
            const float hn = s + bu + L[row] * e;
            hnxt[row] = hn;
            float cp = C[row] * hn;                        // partial C·h_{t+1}
            #pragma unroll
            for (int o = 16; o > 0; o >>= 1) cp += __shfl_down(cp, o, 32);
            if (tid == 0) ypn[wg] = cp;                    // deterministic slot
        }

        // ---- device-wide sense-reversal barrier ---------------------------
        __syncthreads();
        __builtin_amdgcn_fence(__ATOMIC_RELEASE, "agent");
        if (tid == 0) {
            unsigned g0 = __hip_atomic_load(gen, __ATOMIC_RELAXED, __HIP_MEMORY_SCOPE_AGENT);
            unsigned prev = __hip_atomic_fetch_add(cnt, 1u, __ATOMIC_ACQ_REL,
                                                   __HIP_MEMORY_SCOPE_AGENT);
            if (prev == (unsigned)(NWG - 1)) {
                __hip_atomic_store(cnt, 0u, __ATOMIC_RELAXED, __HIP_MEMORY_SCOPE_AGENT);
                __hip_atomic_fetch_add(gen, 1u, __ATOMIC_RELEASE, __HIP_MEMORY_SCOPE_AGENT);
            } else {
                while (__hip_atomic_load(gen, __ATOMIC_ACQUIRE,
                                         __HIP_MEMORY_SCOPE_AGENT) == g0) {
                    __builtin_amdgcn_s_sleep(2);
                }
            }
        }
        __syncthreads();
        __builtin_amdgcn_fence(__ATOMIC_ACQUIRE, "agent");

        // ---- ping-pong ----------------------------------------------------
        float* th = hcur; hcur = hnxt; hnxt = th;
        float* ty = ypc;  ypc  = ypn;  ypn  = ty;
    }
}

__global__ void out_kernel(const Ws* __restrict__ ws, float* __restrict__ out) {
    int t = blockIdx.x * blockDim.x + threadIdx.x;
    if (t < SEQ_LEN) out[t] = tanhf(ws->ybuf[t] + ws->g[t]) * 3.0f;
}

extern "C" void kernel_launch(void* const* d_in, const int* in_sizes, int n_in,
                              void* d_out, int out_size, void* d_ws, size_t ws_size,
                              hipStream_t stream) {
    const float* u    = (const float*)d_in[0];   // [8, 4096]
    const float* yobs = (const float*)d_in[1];   // [1, 4096]
    const float* wa   = (const float*)d_in[2];   // [4096, 4096]
    const float* wb   = (const float*)d_in[3];   // [4096, 8]
    const float* wc   = (const float*)d_in[4];   // [1, 4096]
    const float* wd   = (const float*)d_in[5];   // [1, 8]
    const float* wl   = (const float*)d_in[6];   // [4096, 1]
    float* out = (float*)d_out;
    Ws* ws = (Ws*)d_ws;                          // ~65 KB used

    setup_kernel<<<(ORDER + NTHREADS - 1) / NTHREADS, NTHREADS, 0, stream>>>(u, wd, ws);
    c_sum_kernel<<<1, NTHREADS, 0, stream>>>(wc, ws);
    observer_kernel<<<NWG, NTHREADS, 0, stream>>>(wa, wb, wc, wl, u, yobs, ws);
    out_kernel<<<SEQ_LEN / NTHREADS, NTHREADS, 0, stream>>>(ws, out);
}